// TextGraphMCModel_Att_14379550507193
// MI455X (gfx1250) — compile-verified
//
#include <hip/hip_runtime.h>
#include <math.h>

#define HDIM 128
#define BSEG 256

typedef __attribute__((ext_vector_type(2))) float v2f;
typedef __attribute__((ext_vector_type(8))) float v8f;

// ---------------------------------------------------------------------------
// Kernel 1: ws[b,k] = sum_j s[b,j] * W[k,j]   (ws = s @ W^T)
// WMMA f32 16x16x4: one wave per 16x16 output tile, K-loop over 128 in steps of 4.
// A (16x4 f32): lanes 0-15 / 16-31 both hold M=lane&15; half selects K pair.
// B (4x16 f32): VGPR0 holds rows K=0 (lanes 0-15) / K=2 (lanes 16-31), VGPR1 rows 1/3.
// C/D (16x16 f32): VGPR r -> row r (lanes 0-15) / row r+8 (lanes 16-31), col = lane&15.
// ---------------------------------------------------------------------------
__global__ void __launch_bounds__(32) ws_gemm_wmma(const float* __restrict__ s,
                                                   const float* __restrict__ W,
                                                   float* __restrict__ ws) {
  const int tile = blockIdx.x;          // 16 (B/16) x 8 (H/16) = 128 tiles
  const int b0 = (tile >> 3) * 16;      // row block in B dimension
  const int n0 = (tile & 7) * 16;       // col block in H dimension
  const int lane = threadIdx.x & 31;
  const int l = lane & 15;
  const int half = lane >> 4;

  v8f c = {0.f, 0.f, 0.f, 0.f, 0.f, 0.f, 0.f, 0.f};
  const float* arow = s + (b0 + l) * HDIM;          // A row (s)
  const float* brow = W + (n0 + l) * HDIM;          // B col of W^T = row of W

  for (int k0 = 0; k0 < HDIM; k0 += 4) {
    v2f a, b;
    a.x = arow[k0 + 2 * half];
    a.y = arow[k0 + 2 * half + 1];
    b.x = brow[k0 + 2 * half];
    b.y = brow[k0 + 2 * half + 1];
    c = __builtin_amdgcn_wmma_f32_16x16x4_f32(false, a, false, b,
                                              (short)0, c, false, false);
  }

#pragma unroll
  for (int r = 0; r < 8; ++r)
    ws[(b0 + r + 8 * half) * HDIM + n0 + l] = c[r];
}

// ---------------------------------------------------------------------------
// Kernel 2: h[n] = dot(node_embeds[n], ws[seg[n]])  — one wave32 per node.
// Each lane loads a float4 (coalesced 512B per wave), then shuffle-reduce.
// ---------------------------------------------------------------------------
__global__ void __launch_bounds__(256) score_kernel(const float* __restrict__ x,
                                                    const float* __restrict__ ws,
                                                    const int* __restrict__ seg,
                                                    float* __restrict__ h, int N) {
  const int node = (int)((blockIdx.x * (unsigned)blockDim.x + threadIdx.x) >> 5);
  const int lane = threadIdx.x & 31;
  if (node >= N) return;
  const int b = seg[node];
  const float4 xv = ((const float4*)(x + (size_t)node * HDIM))[lane];
  const float4 wv = ((const float4*)(ws + (size_t)b * HDIM))[lane];
  float d = xv.x * wv.x + xv.y * wv.y + xv.z * wv.z + xv.w * wv.w;
#pragma unroll
  for (int m = 16; m >= 1; m >>= 1) d += __shfl_xor(d, m, 32);
  if (lane == 0) h[node] = d;
}

// ---------------------------------------------------------------------------
// Kernel 3: segment starts via binary search (segment_ids sorted ascending).
// start[b] = first index i with seg[i] >= b, for b in [0, BSEG].
// ---------------------------------------------------------------------------
__global__ void bounds_kernel(const int* __restrict__ seg, int* __restrict__ start,
                              int N) {
  const int b = blockIdx.x * blockDim.x + threadIdx.x;
  if (b > BSEG) return;
  int lo = 0, hi = N;
  while (lo < hi) {
    int mid = (lo + hi) >> 1;
    if (seg[mid] < b) lo = mid + 1; else hi = mid;
  }
  start[b] = lo;
}

// ---------------------------------------------------------------------------
// Kernel 4: per-segment stable softmax + weighted pooling.
// One block (128 threads) per segment: block-max, exp+sum (e cached to global
// scratch), then thread t accumulates feature t across the segment's rows.
// ---------------------------------------------------------------------------
__global__ void __launch_bounds__(128) pool_kernel(const float* __restrict__ x,
                                                   const float* __restrict__ h,
                                                   const int* __restrict__ start,
                                                   float* __restrict__ e,
                                                   float* __restrict__ out) {
  __shared__ float red[128];
  const int b = blockIdx.x;
  const int t = threadIdx.x;
  const int s0 = start[b], s1 = start[b + 1];
  if (s0 >= s1) { out[b * HDIM + t] = 0.f; return; }   // uniform per block

  // --- segment max ---
  float m = -INFINITY;
  for (int i = s0 + t; i < s1; i += 128) m = fmaxf(m, h[i]);
  red[t] = m;
  __syncthreads();
  for (int o = 64; o >= 1; o >>= 1) {
    if (t < o) red[t] = fmaxf(red[t], red[t + o]);
    __syncthreads();
  }
  m = red[0];
  __syncthreads();

  // --- exp + segment sum (cache e[i] in scratch) ---
  float sum = 0.f;
  for (int i = s0 + t; i < s1; i += 128) {
    const float ev = __expf(h[i] - m);
    e[i] = ev;
    sum += ev;
  }
  red[t] = sum;
  __syncthreads();
  for (int o = 64; o >= 1; o >>= 1) {
    if (t < o) red[t] += red[t + o];
    __syncthreads();
  }
  const float inv = 1.0f / red[0];

  // --- weighted pooling: thread t owns feature t ---
  float acc = 0.f;
  for (int i = s0; i < s1; ++i)
    acc += e[i] * x[(size_t)i * HDIM + t];
  out[b * HDIM + t] = acc * inv;
}

// ---------------------------------------------------------------------------
extern "C" void kernel_launch(void* const* d_in, const int* in_sizes, int n_in,
                              void* d_out, int out_size, void* d_ws, size_t ws_size,
                              hipStream_t stream) {
  const float* s   = (const float*)d_in[0];  // [B, H]
  const float* x   = (const float*)d_in[1];  // [N, H] node_embeds
  const float* W   = (const float*)d_in[2];  // [H, H]
  const int*   seg = (const int*)d_in[3];    // [N]
  const int N = in_sizes[3];
  float* out = (float*)d_out;                // [B, H]

  char* wsb = (char*)d_ws;
  float* ws    = (float*)wsb;                                        // B*H floats
  float* h     = (float*)(wsb + (size_t)BSEG * HDIM * 4);            // N floats
  float* e     = (float*)(wsb + ((size_t)BSEG * HDIM + N) * 4);      // N floats
  int*   start = (int*)  (wsb + ((size_t)BSEG * HDIM + 2ull * N) * 4); // B+1 ints

  ws_gemm_wmma<<<(BSEG / 16) * (HDIM / 16), 32, 0, stream>>>(s, W, ws);
  score_kernel<<<(N + 7) / 8, 256, 0, stream>>>(x, ws, seg, h, N);
  bounds_kernel<<<1, 288, 0, stream>>>(seg, start, N);
  pool_kernel<<<BSEG, 128, 0, stream>>>(x, h, start, e, out);
}